// FastViewTransformer_65841848647984
// MI455X (gfx1250) — compile-verified
//
#include <hip/hip_runtime.h>
#include <hip/hip_bf16.h>
#include <stdint.h>

typedef float        v2f  __attribute__((ext_vector_type(2)));
typedef float        v8f  __attribute__((ext_vector_type(8)));
typedef unsigned int v4u  __attribute__((ext_vector_type(4)));
typedef int          v4i  __attribute__((ext_vector_type(4)));
typedef int          v8i  __attribute__((ext_vector_type(8)));
typedef unsigned int u32;
typedef unsigned long long u64;

#define B_    2
#define C_    80
#define H_    32
#define W_    88
#define HW_   (H_ * W_)          // 2816
#define NPTS  262144             // 128*128*16 frustum points per batch
#define CT    8                  // channels per LDS tile
#define NTIL  (C_ / CT)          // 10 channel tiles
#define TPB   256
#define PPT   8                  // points per thread per pass
#define PT    (TPB * PPT)        // 2048 points per block

// ---------------------------------------------------------------------------
// Kernel 1: per-batch combined projection matrix  M = (K*s) @ E @ bda  (3x4)
// computed with two V_WMMA_F32_16X16X4_F32 ops (the only matrix ops here).
// One wave (32 lanes).  A(16x4): lane m=lane&15 holds K=2g+j (g=lane>>4) in
// v2f; B(4x16): lane n=lane&15 holds rows K=2g+j; D(16x16): d[v] = row v+8g.
// ---------------------------------------------------------------------------
__global__ __launch_bounds__(32)
void fvt_proj_wmma(const float* __restrict__ intrin,
                   const float* __restrict__ extrin,
                   const float* __restrict__ bda,
                   float* __restrict__ Mout) {
  const int lane = threadIdx.x;
  const int m = lane & 15;
  const int g = lane >> 4;
  const int k0 = 2 * g, k1 = 2 * g + 1;
  __shared__ float P[3][4];

  for (int b = 0; b < B_; ++b) {
    const float* Kb = intrin + b * 16;   // (1,4,4) -> 4x4
    const float* Eb = extrin + b * 16;
    const float* Db = bda + b * 16;

    // ---- WMMA #1 : P(3x4) = S(3x3, row-scaled) @ E(3x4), K-dim padded to 4
    const float rs = (m < 2) ? 0.125f : 1.0f;   // rows 0,1 scaled by 1/DS
    v2f a1;
    a1.x = (m < 3 && k0 < 3) ? Kb[m * 4 + k0] * rs : 0.0f;
    a1.y = (m < 3 && k1 < 3) ? Kb[m * 4 + k1] * rs : 0.0f;
    v2f b1;
    b1.x = (k0 < 3 && m < 4) ? Eb[k0 * 4 + m] : 0.0f;
    b1.y = (k1 < 3 && m < 4) ? Eb[k1 * 4 + m] : 0.0f;
    v8f c0 = {};
    v8f d1 = __builtin_amdgcn_wmma_f32_16x16x4_f32(
        false, a1, false, b1, (short)0, c0, false, false);
    if (lane < 4) {           // lanes 0..3 hold cols 0..3 of rows 0..2
      P[0][lane] = d1[0];
      P[1][lane] = d1[1];
      P[2][lane] = d1[2];
    }
    __syncthreads();

    // ---- WMMA #2 : M(3x4) = P(3x4) @ bda(4x4)
    v2f a2;
    a2.x = (m < 3) ? P[m][k0] : 0.0f;
    a2.y = (m < 3) ? P[m][k1] : 0.0f;
    v2f b2;
    b2.x = (m < 4) ? Db[k0 * 4 + m] : 0.0f;
    b2.y = (m < 4) ? Db[k1 * 4 + m] : 0.0f;
    v8f d2 = __builtin_amdgcn_wmma_f32_16x16x4_f32(
        false, a2, false, b2, (short)0, c0, false, false);
    if (lane < 4) {
      Mout[b * 12 + 0 + lane] = d2[0];
      Mout[b * 12 + 4 + lane] = d2[1];
      Mout[b * 12 + 8 + lane] = d2[2];
    }
    __syncthreads();
  }
}

// ---------------------------------------------------------------------------
// TDM: stage one contiguous 8-channel feature slab (CT*HW_ f32 = 90112 B)
// global -> LDS via tensor_load_to_lds.  Descriptor per CDNA5 ISA §8.3-8.6:
// 1-row 2D tile, data_size=4B, no pad/iterate/multicast.
// ---------------------------------------------------------------------------
__device__ __forceinline__
void tdm_stage_slab(u32 lds_byte_off, const float* gsrc) {
  const u64 ga = (u64)(uintptr_t)gsrc;
  const u32 nelem = (u32)(CT * HW_);     // 22528 elements (fits tile_dim0 u16)

  v4u g0;
  g0.x = 1u;                                            // count=1, user D#
  g0.y = lds_byte_off;                                  // lds_addr
  g0.z = (u32)(ga & 0xffffffffu);                       // global_addr[31:0]
  g0.w = (u32)((ga >> 32) & 0x01ffffffu) | (2u << 30);  // addr[56:32] | type=2

  v8i g1;
  g1[0] = (int)(2u << 16);                        // data_size = 2 -> 4 bytes
  g1[1] = (int)((nelem & 0xffffu) << 16);         // tensor_dim0[15:0]
  g1[2] = (int)((nelem >> 16) | (1u << 16));      // tensor_dim0[31:16] | tensor_dim1=1
  g1[3] = (int)((nelem & 0xffffu) << 16);         // tensor_dim1 hi=0 | tile_dim0
  g1[4] = (int)1;                                 // tile_dim1=1, tile_dim2=0
  g1[5] = (int)nelem;                             // tensor_dim0_stride[31:0]
  g1[6] = 0;                                      // stride hi | dim1_stride lo
  g1[7] = 0;                                      // dim1_stride hi

  v4i gz = {0, 0, 0, 0};
#if __clang_major__ >= 23
  v8i gz8 = {0, 0, 0, 0, 0, 0, 0, 0};
  __builtin_amdgcn_tensor_load_to_lds(g0, g1, gz, gz, gz8, 0);
#else
  __builtin_amdgcn_tensor_load_to_lds(g0, g1, gz, gz, 0);
#endif
}

// ---------------------------------------------------------------------------
// Kernel 2: project 2048 points/block, bilinear-gather 80 channels from
// TDM-staged, double-buffered LDS slabs; coalesced non-temporal stores.
// ---------------------------------------------------------------------------
__global__ __launch_bounds__(TPB)
void fvt_bev_kernel(const float* __restrict__ feats,
                    const float* __restrict__ Mmat,
                    float* __restrict__ out) {
  __shared__ float lds[2][CT * HW_];     // 2 x 90112 B = 180224 B (<320 KB WGP)

  const int b   = blockIdx.y;
  const int n0  = blockIdx.x * PT;
  const int tid = threadIdx.x;

  float M[12];
#pragma unroll
  for (int i = 0; i < 12; ++i) M[i] = Mmat[b * 12 + i];

  // ---- per-point bilinear indices & weights, kept in registers across tiles
  u32   gi[PPT][4];
  float gw[PPT][4];
#pragma unroll
  for (int i = 0; i < PPT; ++i) {
    const int n  = n0 + i * TPB + tid;
    const int iz = n & 15;
    const int iy = (n >> 4) & 127;
    const int ix = n >> 11;
    const float x = -51.2f + 0.8f * (float)ix;
    const float y = -51.2f + 0.8f * (float)iy;
    const float z = -5.0f + 0.5f * (float)iz;
    const float px = M[0] * x + M[1] * y + M[2]  * z + M[3];
    const float py = M[4] * x + M[5] * y + M[6]  * z + M[7];
    const float pz = M[8] * x + M[9] * y + M[10] * z + M[11];
    float xr = (px / pz) * ((float)(W_ - 1) / (float)W_);
    float yr = (py / pz) * ((float)(H_ - 1) / (float)H_);
    if (!(isfinite(xr) && isfinite(yr))) { xr = -10.0f; yr = -10.0f; }
    const float x0f = floorf(xr), y0f = floorf(yr);
    const float wx1 = xr - x0f,  wy1 = yr - y0f;
    const float wx0 = 1.0f - wx1, wy0 = 1.0f - wy1;
    const int x0 = (int)x0f, y0 = (int)y0f;
    const int x1 = x0 + 1,   y1 = y0 + 1;
    const float vx0 = (x0 >= 0 && x0 < W_) ? 1.0f : 0.0f;
    const float vx1 = (x1 >= 0 && x1 < W_) ? 1.0f : 0.0f;
    const float vy0 = (y0 >= 0 && y0 < H_) ? 1.0f : 0.0f;
    const float vy1 = (y1 >= 0 && y1 < H_) ? 1.0f : 0.0f;
    const int cx0 = x0 < 0 ? 0 : (x0 > W_ - 1 ? W_ - 1 : x0);
    const int cx1 = x1 < 0 ? 0 : (x1 > W_ - 1 ? W_ - 1 : x1);
    const int cy0 = y0 < 0 ? 0 : (y0 > H_ - 1 ? H_ - 1 : y0);
    const int cy1 = y1 < 0 ? 0 : (y1 > H_ - 1 ? H_ - 1 : y1);
    gi[i][0] = (u32)(cy0 * W_ + cx0);  gw[i][0] = wx0 * wy0 * vx0 * vy0;
    gi[i][1] = (u32)(cy0 * W_ + cx1);  gw[i][1] = wx1 * wy0 * vx1 * vy0;
    gi[i][2] = (u32)(cy1 * W_ + cx0);  gw[i][2] = wx0 * wy1 * vx0 * vy1;
    gi[i][3] = (u32)(cy1 * W_ + cx1);  gw[i][3] = wx1 * wy1 * vx1 * vy1;
  }

  const float* fb = feats + (size_t)b * C_ * HW_;

  // ---- prologue: TDM-stage tile 0
  if (tid == 0) tdm_stage_slab((u32)(uintptr_t)&lds[0][0], fb);

  for (int t = 0; t < NTIL; ++t) {
    if (tid == 0) {
      if (t + 1 < NTIL) {
        tdm_stage_slab((u32)(uintptr_t)&lds[(t + 1) & 1][0],
                       fb + (size_t)(t + 1) * CT * HW_);
        __builtin_amdgcn_s_wait_tensorcnt(1);   // tile t done (in-order TDM)
      } else {
        __builtin_amdgcn_s_wait_tensorcnt(0);
      }
    }
    __syncthreads();

    const float* L  = &lds[t & 1][0];
    float*       ob = out + ((size_t)b * C_ + (size_t)t * CT) * NPTS + n0;
#pragma unroll
    for (int i = 0; i < PPT; ++i) {
      const int   off = i * TPB + tid;
      const float w0 = gw[i][0], w1 = gw[i][1], w2 = gw[i][2], w3 = gw[i][3];
      const u32   i0 = gi[i][0], i1 = gi[i][1], i2 = gi[i][2], i3 = gi[i][3];
#pragma unroll
      for (int c = 0; c < CT; ++c) {
        const float* Lc = L + c * HW_;
        const float v = w0 * Lc[i0] + w1 * Lc[i1] + w2 * Lc[i2] + w3 * Lc[i3];
        __builtin_nontemporal_store(v, &ob[(size_t)c * NPTS + off]);
      }
    }
    __syncthreads();   // all reads of buf (t&1) done before it is re-staged
  }
}

// ---------------------------------------------------------------------------
extern "C" void kernel_launch(void* const* d_in, const int* in_sizes, int n_in,
                              void* d_out, int out_size, void* d_ws, size_t ws_size,
                              hipStream_t stream) {
  const float* feats  = (const float*)d_in[0];   // (2,80,32,88) f32
  const float* intrin = (const float*)d_in[1];   // (2,1,4,4)
  const float* extrin = (const float*)d_in[2];   // (2,1,4,4)
  const float* bda    = (const float*)d_in[3];   // (2,4,4)
  float* out = (float*)d_out;                    // (2,80,128,128,16) f32
  float* Mws = (float*)d_ws;                     // 24 floats scratch

  fvt_proj_wmma<<<dim3(1), dim3(32), 0, stream>>>(intrin, extrin, bda, Mws);
  fvt_bev_kernel<<<dim3(NPTS / PT, B_), dim3(TPB), 0, stream>>>(feats, Mws, out);
}